// TopologyLayer_34239479283880
// MI455X (gfx1250) — compile-verified
//
#include <hip/hip_runtime.h>
#include <math.h>

// ---------------- problem constants (from reference) ----------------
#define N_NODES 20000
#define N_EDGES 100000
#define F_IN    1024
#define F_OUT   512
#define NUM_FILT 8
#define COORD_F (NUM_FILT * 12)      // 96
#define K_TOT   (F_IN + COORD_F)     // 1120
#define CAP     131072               // pow2 >= N_EDGES
#define CAP_LOG 17
#define BCHUNK  2048                 // LDS bitonic chunk (16 KB of u64)

typedef float v2f __attribute__((ext_vector_type(2)));
typedef float v8f __attribute__((ext_vector_type(8)));
typedef unsigned su4 __attribute__((ext_vector_type(4)));
typedef unsigned su8 __attribute__((ext_vector_type(8)));
typedef unsigned long long u64;
typedef unsigned int u32;

// =====================================================================
// Kernel 1: f_all[j][i] = dot(x[i], filt_w[j]) + filt_b[j]
// wave-per-node, filt_w cached in LDS (32 KB). Memory-bound (reads x once).
// =====================================================================
__global__ __launch_bounds__(256) void filt_gemm_kernel(
    const float* __restrict__ x, const float* __restrict__ w,
    const float* __restrict__ b, float* __restrict__ f_all) {
  __shared__ float ws[NUM_FILT * F_IN];
  const int tid = threadIdx.x;
#pragma unroll
  for (int q = 0; q < 8; ++q) {
    int f4 = tid + q * 256;  // 2048 float4 = 8192 floats
    ((float4*)ws)[f4] = ((const float4*)w)[f4];
  }
  __syncthreads();
  const int wave = tid >> 5, lane = tid & 31;
  const int node = blockIdx.x * 8 + wave;  // 20000 = 2500*8, exact
  float acc[NUM_FILT];
#pragma unroll
  for (int j = 0; j < NUM_FILT; ++j) acc[j] = 0.f;
  const float4* xr = (const float4*)(x + (size_t)node * F_IN);
#pragma unroll
  for (int c = 0; c < 8; ++c) {
    const int k4 = c * 32 + lane;
    const float4 xv = xr[k4];
#pragma unroll
    for (int j = 0; j < NUM_FILT; ++j) {
      const float4 wv = ((const float4*)ws)[j * 256 + k4];
      acc[j] += xv.x * wv.x + xv.y * wv.y + xv.z * wv.z + xv.w * wv.w;
    }
  }
#pragma unroll
  for (int j = 0; j < NUM_FILT; ++j) {
    float r = acc[j];
#pragma unroll
    for (int off = 16; off > 0; off >>= 1) r += __shfl_down(r, off, 32);
    if (lane == 0) f_all[j * N_NODES + node] = r + b[j];
  }
}

// =====================================================================
// Kernel 2: 64-bit sort keys (sign-flipped f32 << 32) | edge_idx.
// Low-bit index == stable argsort tiebreak (matches jnp.argsort).
// =====================================================================
__device__ __forceinline__ u32 f32_ordered(float m) {
  u32 bb = __float_as_uint(m);
  return (bb & 0x80000000u) ? ~bb : (bb | 0x80000000u);
}

__global__ __launch_bounds__(256) void key_kernel(
    const float* __restrict__ f_all, const int* __restrict__ ei,
    u64* __restrict__ buf) {
  const int t = blockIdx.x * 256 + threadIdx.x;
  if (t >= NUM_FILT * CAP) return;
  const int j = t >> CAP_LOG;
  const int e = t & (CAP - 1);
  u64 val;
  if (e < N_EDGES) {
    const int u = ei[e], v = ei[N_EDGES + e];
    const float fu = f_all[j * N_NODES + u];
    const float fv = f_all[j * N_NODES + v];
    val = ((u64)f32_ordered(fmaxf(fu, fv)) << 32) | (u32)e;
  } else {
    val = ~0ull;
  }
  buf[(size_t)j * CAP + e] = val;
}

// =====================================================================
// Kernel 3a: LDS presort — all sizes 2..2048 inside a 2048-elem chunk.
// =====================================================================
__global__ __launch_bounds__(256) void bitonic_local_presort(u64* __restrict__ buf) {
  __shared__ u64 sh[BCHUNK];
  const size_t base = (size_t)blockIdx.x * BCHUNK;
#pragma unroll
  for (int q = 0; q < 8; ++q)
    sh[threadIdx.x + q * 256] = buf[base + threadIdx.x + q * 256];
  __syncthreads();
  const unsigned chunkStart = (unsigned)(base & (CAP - 1));
  for (int size = 2; size <= BCHUNK; size <<= 1) {
    for (int stride = size >> 1; stride >= 1; stride >>= 1) {
#pragma unroll
      for (int q = 0; q < 4; ++q) {
        const int p = threadIdx.x + q * 256;  // pair 0..1023
        const int i = 2 * p - (p & (stride - 1));
        const bool up = (((chunkStart + (unsigned)i) & (unsigned)size) == 0);
        const u64 a = sh[i], b = sh[i + stride];
        if ((a > b) == up) { sh[i] = b; sh[i + stride] = a; }
      }
      __syncthreads();
    }
  }
#pragma unroll
  for (int q = 0; q < 8; ++q)
    buf[base + threadIdx.x + q * 256] = sh[threadIdx.x + q * 256];
}

// =====================================================================
// Kernel 3b: one global compare-exchange step (stride >= 2048).
// =====================================================================
__global__ __launch_bounds__(256) void bitonic_step(
    u64* __restrict__ buf, int size, int stride) {
  const int t = blockIdx.x * 256 + threadIdx.x;
  if (t >= NUM_FILT * (CAP / 2)) return;
  const int j = t >> (CAP_LOG - 1);
  const int r = t & (CAP / 2 - 1);
  const int i = 2 * r - (r & (stride - 1));
  u64* p = buf + (size_t)j * CAP;
  const u64 a = p[i];
  const u64 b = p[i + stride];
  const bool up = ((i & size) == 0);
  if ((a > b) == up) { p[i] = b; p[i + stride] = a; }
}

// =====================================================================
// Kernel 3c: finish a size-stage in LDS (strides 1024..1).
// =====================================================================
__global__ __launch_bounds__(256) void bitonic_local_finish(
    u64* __restrict__ buf, int size) {
  __shared__ u64 sh[BCHUNK];
  const size_t base = (size_t)blockIdx.x * BCHUNK;
#pragma unroll
  for (int q = 0; q < 8; ++q)
    sh[threadIdx.x + q * 256] = buf[base + threadIdx.x + q * 256];
  __syncthreads();
  const unsigned chunkStart = (unsigned)(base & (CAP - 1));
  const bool up = ((chunkStart & (unsigned)size) == 0);
  for (int stride = BCHUNK / 2; stride >= 1; stride >>= 1) {
#pragma unroll
    for (int q = 0; q < 4; ++q) {
      const int p = threadIdx.x + q * 256;
      const int i = 2 * p - (p & (stride - 1));
      const u64 a = sh[i], b = sh[i + stride];
      if ((a > b) == up) { sh[i] = b; sh[i + stride] = a; }
    }
    __syncthreads();
  }
#pragma unroll
  for (int q = 0; q < 8; ++q)
    buf[base + threadIdx.x + q * 256] = sh[threadIdx.x + q * 256];
}

// =====================================================================
// Kernel 4: serial union-find per filtration, state fully in LDS.
// parent + death + f = 240 KB (fits CDNA5's 320 KB WGP LDS).
// =====================================================================
__global__ __launch_bounds__(256) void union_find_kernel(
    const float* __restrict__ f_all, const u64* __restrict__ sorted,
    const int* __restrict__ ei, int* __restrict__ death_g) {
  extern __shared__ char smem[];
  u32* parent = (u32*)smem;
  u32* death  = parent + N_NODES;
  float* fv   = (float*)(death + N_NODES);
  const int j = blockIdx.x;
  const float* f = f_all + (size_t)j * N_NODES;
  for (int i = threadIdx.x; i < N_NODES; i += 256) {
    parent[i] = i; death[i] = i; fv[i] = f[i];
  }
  __syncthreads();
  if (threadIdx.x == 0) {
    const u64* s = sorted + (size_t)j * CAP;
    for (int e = 0; e < N_EDGES; ++e) {
      const u32 idx = (u32)(s[e] & 0xFFFFFFFFu);
      const int u = ei[idx], v = ei[N_EDGES + idx];
      const float fu = fv[u], fw = fv[v];
      const u32 d = (fu >= fw) ? (u32)u : (u32)v;
      int a = u;
      for (;;) { u32 pa = parent[a]; if ((int)pa == a) break;
                 u32 ga = parent[pa]; parent[a] = ga; a = (int)ga; }
      int c = v;
      for (;;) { u32 pc = parent[c]; if ((int)pc == c) break;
                 u32 gc = parent[pc]; parent[c] = gc; c = (int)gc; }
      if (a != c) {
        const int older = (fv[a] <= fv[c]) ? a : c;
        const int younger = a + c - older;
        parent[younger] = (u32)older;
        death[younger] = d;
      }
    }
  }
  __syncthreads();
  for (int i = threadIdx.x; i < N_NODES; i += 256)
    death_g[j * N_NODES + i] = (int)death[i];
}

// =====================================================================
// Kernel 5: coordinate-function features -> coord[N][96]
// =====================================================================
__global__ __launch_bounds__(256) void coord_kernel(
    const float* __restrict__ f_all, const int* __restrict__ death,
    const float* __restrict__ tri_t, const float* __restrict__ gauss_mu,
    const float* __restrict__ gauss_sigma, const float* __restrict__ line_w,
    const float* __restrict__ rhat_c, const float* __restrict__ rhat_r,
    float* __restrict__ coord) {
  const int i = blockIdx.x * 256 + threadIdx.x;
  if (i >= N_NODES) return;
  const float sig = gauss_sigma[0];
  const float inv2s2 = 1.f / (2.f * sig * sig);
  const float rr = fabsf(rhat_r[0]);
  for (int j = 0; j < NUM_FILT; ++j) {
    const float b = f_all[j * N_NODES + i];
    const int di = death[j * N_NODES + i];
    const float d = f_all[j * N_NODES + di];
    float* o = coord + (size_t)i * COORD_F + j * 12;
#pragma unroll
    for (int t = 0; t < 3; ++t)
      o[t] = fmaxf(d - fabsf(tri_t[t] - b), 0.f);
#pragma unroll
    for (int t = 0; t < 3; ++t) {
      const float dx = b - gauss_mu[t * 2 + 0];
      const float dy = d - gauss_mu[t * 2 + 1];
      o[3 + t] = expf(-(dx * dx + dy * dy) * inv2s2);
    }
#pragma unroll
    for (int t = 0; t < 3; ++t)
      o[6 + t] = b * line_w[t * 2 + 0] + d * line_w[t * 2 + 1];
#pragma unroll
    for (int t = 0; t < 3; ++t) {
      const float q = fabsf(b - rhat_c[t * 2 + 0]) + fabsf(d - rhat_c[t * 2 + 1]);
      o[9 + t] = 1.f / (1.f + q) - 1.f / (1.f + fabsf(rr - q));
    }
  }
}

// =====================================================================
// Kernel 6: out[N][512] = [x | coord] @ out_w^T + out_b  via
// V_WMMA_F32_16X16X4_F32, software-pipelined double-buffered tiles:
//  - A tile: per-lane GLOBAL_LOAD_ASYNC_TO_LDS_B128, ASYNCcnt partial waits
//    (4 ops/wave/chunk; in-order completion -> wait asynccnt<=4 keeps the
//    next chunk in flight while the current one computes).
//  - B tile: Tensor Data Mover, TENSORcnt partial waits (<=1). TDM pad
//    (32 dw + 4 dw) reproduces the conflict-free LDS row stride 36.
// =====================================================================
#define MT 128
#define NT 64
#define KT 32
#define ASTR 36

__global__ __launch_bounds__(256) void out_gemm_kernel(
    const float* __restrict__ x, const float* __restrict__ coord,
    const float* __restrict__ w,   // [F_OUT][K_TOT]
    const float* __restrict__ bias, float* __restrict__ out) {
  __shared__ float As[2][MT * ASTR];  // 2 x 18 KB
  __shared__ float Bs[2][NT * ASTR];  // 2 x  9 KB
  const int tid = threadIdx.x;
  const int lane = tid & 31, wave = tid >> 5;
  const int mw = wave & 3, nw = wave >> 2;   // 4x2 wave grid over 128x64
  const int lm = lane & 15, lh = lane >> 4;
  const int i0 = blockIdx.x * MT;
  const int o0 = blockIdx.y * NT;

  v8f acc[2][2];
#pragma unroll
  for (int tm = 0; tm < 2; ++tm)
#pragma unroll
    for (int tn = 0; tn < 2; ++tn)
#pragma unroll
      for (int v = 0; v < 8; ++v) acc[tm][tn][v] = 0.f;

  // ---- TDM descriptor group 1 (constant across chunks) ----
  su8 g1;
  g1[0] = (2u << 16) | (1u << 20) | (4u << 22) | (3u << 25);
  g1[1] = ((u32)K_TOT & 0xFFFFu) << 16;                 // tensor_dim0 lo
  g1[2] = ((u32)K_TOT >> 16) | (((u32)F_OUT & 0xFFFFu) << 16);
  g1[3] = ((u32)F_OUT >> 16) | ((u32)KT << 16);         // tile_dim0
  g1[4] = (u32)NT;                                      // tile_dim1
  g1[5] = (u32)K_TOT;                                   // dim0_stride lo
  g1[6] = 0u;
  g1[7] = 0u;
  const unsigned asBase0 = (unsigned)(size_t)(void*)&As[0][0];
  const unsigned asBase1 = (unsigned)(size_t)(void*)&As[1][0];
  const unsigned bsBase0 = (unsigned)(size_t)(void*)&Bs[0][0];
  const unsigned bsBase1 = (unsigned)(size_t)(void*)&Bs[1][0];
  const u64 wTileBase = (u64)(size_t)(w + (size_t)o0 * K_TOT);

  const int nChunks = K_TOT / KT;  // 35: 0..31 from x, 32..34 from coord

  // issue tiles for chunk chIdx into LDS buffer `buf`
  auto issueTiles = [&](int chIdx, int buf) {
    const int k0 = chIdx * KT;
    // B tile via TDM (one wave per workgroup)
    if (wave == 0) {
      const u64 ga = wTileBase + (u64)k0 * 4u;
      su4 g0;
      g0[0] = 1u;                                       // count=1, user desc
      g0[1] = buf ? bsBase1 : bsBase0;                  // lds_addr
      g0[2] = (u32)ga;                                  // global_addr lo
      g0[3] = (u32)((ga >> 32) & 0x01FFFFFFu) | 0x80000000u;  // hi | type=2
      asm volatile("tensor_load_to_lds %0, %1" :: "s"(g0), "s"(g1) : "memory");
    }
    // A tile via per-lane async copies: 4 b128 instructions per wave
    const float* src;
    int stride, koff;
    if (k0 < F_IN) { src = x;     stride = F_IN;    koff = k0; }
    else           { src = coord; stride = COORD_F; koff = k0 - F_IN; }
    const unsigned aB = buf ? asBase1 : asBase0;
#pragma unroll
    for (int q = 0; q < 4; ++q) {
      const int id = tid + q * 256;
      const int r = id >> 3, c4 = id & 7;
      int row = i0 + r;
      if (row >= N_NODES) row = N_NODES - 1;  // clamp: masked at store
      const float* gp = src + (size_t)row * stride + koff + c4 * 4;
      const unsigned ldsOff = aB + (unsigned)((r * ASTR + c4 * 4) * 4);
      asm volatile("global_load_async_to_lds_b128 %0, %1, off"
                   :: "v"(ldsOff), "v"((u64)(size_t)gp) : "memory");
    }
  };

  issueTiles(0, 0);  // prologue

  for (int ch = 0; ch < nChunks; ++ch) {
    const int cur = ch & 1;
    // keep the pipeline full: start chunk ch+1 into the other buffer.
    // Its last readers (chunk ch-1) were released by the trailing barrier
    // of the previous iteration.
    if (ch + 1 < nChunks) {
      issueTiles(ch + 1, cur ^ 1);
      // require only the OLDER chunk complete (in-order per wave):
      asm volatile("s_wait_asynccnt 0x4" ::: "memory");
      if (wave == 0) __builtin_amdgcn_s_wait_tensorcnt(1);
    } else {
      asm volatile("s_wait_asynccnt 0x0" ::: "memory");
      if (wave == 0) __builtin_amdgcn_s_wait_tensorcnt(0);
    }
    __syncthreads();  // publish buffer `cur` to all waves

    const float* Ac = As[cur];
    const float* Bc = Bs[cur];
#pragma unroll
    for (int kk = 0; kk < KT; kk += 4) {
      v2f a[2], bfrag[2];
#pragma unroll
      for (int tm = 0; tm < 2; ++tm) {
        const int r = mw * 32 + tm * 16 + lm;
        a[tm].x = Ac[r * ASTR + kk + lh * 2];
        a[tm].y = Ac[r * ASTR + kk + lh * 2 + 1];
      }
#pragma unroll
      for (int tn = 0; tn < 2; ++tn) {
        const int r = nw * 32 + tn * 16 + lm;
        bfrag[tn].x = Bc[r * ASTR + kk + lh * 2];
        bfrag[tn].y = Bc[r * ASTR + kk + lh * 2 + 1];
      }
#pragma unroll
      for (int tm = 0; tm < 2; ++tm)
#pragma unroll
        for (int tn = 0; tn < 2; ++tn)
          acc[tm][tn] = __builtin_amdgcn_wmma_f32_16x16x4_f32(
              false, a[tm], false, bfrag[tn], (short)0, acc[tm][tn],
              false, false);
    }
    __syncthreads();  // release buffer `cur` for the chunk ch+2 writes
  }

  // epilogue: add bias, store (guard partial last M block)
#pragma unroll
  for (int tm = 0; tm < 2; ++tm) {
    const int rbase = i0 + mw * 32 + tm * 16 + lh * 8;
#pragma unroll
    for (int tn = 0; tn < 2; ++tn) {
      const int col = o0 + nw * 32 + tn * 16 + lm;
      const float bc = bias[col];
#pragma unroll
      for (int v = 0; v < 8; ++v) {
        const int row = rbase + v;
        if (row < N_NODES) out[(size_t)row * F_OUT + col] = acc[tm][tn][v] + bc;
      }
    }
  }
}

// =====================================================================
// launch
// =====================================================================
extern "C" void kernel_launch(void* const* d_in, const int* in_sizes, int n_in,
                              void* d_out, int out_size, void* d_ws, size_t ws_size,
                              hipStream_t stream) {
  const float* x          = (const float*)d_in[0];
  const int*   ei         = (const int*)  d_in[1];
  const float* filt_w     = (const float*)d_in[2];
  const float* filt_b     = (const float*)d_in[3];
  const float* tri_t      = (const float*)d_in[4];
  const float* gauss_mu   = (const float*)d_in[5];
  const float* gauss_sig  = (const float*)d_in[6];
  const float* line_w     = (const float*)d_in[7];
  const float* rhat_c     = (const float*)d_in[8];
  const float* rhat_r     = (const float*)d_in[9];
  const float* out_w      = (const float*)d_in[10];
  const float* out_b      = (const float*)d_in[11];
  float* out = (float*)d_out;

  // workspace layout (bytes):
  //   f_all   : 8*20000*4   =   640,000   @ 0
  //   death   : 8*20000*4   =   640,000   @ 640,000
  //   coord   : 20000*96*4  = 7,680,000   @ 1,280,000
  //   sortbuf : 8*131072*8  = 8,388,608   @ 8,960,000   (total ~16.6 MB)
  char* ws = (char*)d_ws;
  float* f_all   = (float*)(ws);
  int*   death   = (int*)  (ws + 640000);
  float* coord   = (float*)(ws + 1280000);
  u64*   sortbuf = (u64*)  (ws + 8960000);

  filt_gemm_kernel<<<N_NODES / 8, 256, 0, stream>>>(x, filt_w, filt_b, f_all);

  key_kernel<<<(NUM_FILT * CAP) / 256, 256, 0, stream>>>(f_all, ei, sortbuf);

  // hybrid bitonic sort: 28 launches instead of 153
  const int nChunksSort = NUM_FILT * CAP / BCHUNK;  // 512
  bitonic_local_presort<<<nChunksSort, 256, 0, stream>>>(sortbuf);
  for (int size = 2 * BCHUNK; size <= CAP; size <<= 1) {
    for (int stride = size >> 1; stride >= BCHUNK; stride >>= 1)
      bitonic_step<<<(NUM_FILT * (CAP / 2)) / 256, 256, 0, stream>>>(sortbuf, size, stride);
    bitonic_local_finish<<<nChunksSort, 256, 0, stream>>>(sortbuf, size);
  }

  union_find_kernel<<<NUM_FILT, 256, 3 * N_NODES * 4, stream>>>(f_all, sortbuf, ei, death);

  coord_kernel<<<(N_NODES + 255) / 256, 256, 0, stream>>>(
      f_all, death, tri_t, gauss_mu, gauss_sig, line_w, rhat_c, rhat_r, coord);

  dim3 grid((N_NODES + MT - 1) / MT, F_OUT / NT);  // 157 x 8
  out_gemm_kernel<<<grid, 256, 0, stream>>>(x, coord, out_w, out_b, out);
}